// ButterflyModule_79233556676747
// MI455X (gfx1250) — compile-verified
//
#include <hip/hip_runtime.h>
#include <hip/hip_bf16.h>

// CDNA5 / gfx1250 butterfly operator.
//
// Per 16-row block b: z = M_out[b] * smooth_relu(M_in[b] * x + bias[b]),
// x is 16 x B.  M_in/M_out are the composed 4-layer butterfly rotations
// (built once by a tiny kernel), and the main kernel streams the 128MB
// matrix through V_WMMA_F32_16X16X4_F32 at full f32 precision.
// Bandwidth-bound: 256 MB traffic -> ~11us floor at 23.3 TB/s; this round
// trims VALU (fast v_sqrt_f32, immediate-offset addressing) and doubles
// per-lane memory access width (b64) so VALU/vmem issue stays under the
// HBM roofline.

typedef float v2f __attribute__((ext_vector_type(2)));
typedef float v8f __attribute__((ext_vector_type(8)));

#define COL_BLOCK 16
#define NUM_ACT 8
#define CURV2 0.01f   // CURVATURE^2 = 0.1^2

// ---------------------------------------------------------------------------
// Kernel 1: build per-block composed 16x16 matrices.
//   lanes 0..15  : columns of M_in  (layers 0..3, strides 1,2,4,8)
//   lanes 16..31 : columns of M_out (layers 4..7, strides 1,2,4,8)
// ws layout: mats[b][t][r][k] row-major (t=0 -> M_in, t=1 -> M_out).
// ---------------------------------------------------------------------------
__global__ void butterfly_build_mats(const float* __restrict__ angles,
                                     float* __restrict__ mats, int nb) {
  const int b    = blockIdx.x;
  const int lane = threadIdx.x;      // 0..31
  const int t    = lane >> 4;        // 0: input stack, 1: output stack
  const int j    = lane & 15;        // column of the 16x16 matrix this lane owns

  float m[16];
#pragma unroll
  for (int r = 0; r < 16; ++r) m[r] = (r == j) ? 1.0f : 0.0f;

  const int half_n = nb * 8;         // angles row length = N/2

#pragma unroll
  for (int l = 0; l < 4; ++l) {
    const int s = 1 << l;            // stride (compile-time after unroll)
    const float* ang = angles + (size_t)(t * 4 + l) * half_n + (size_t)b * 8;
#pragma unroll
    for (int p = 0; p < 8; ++p) {
      const int rlo = ((p & ~(s - 1)) << 1) | (p & (s - 1));
      const int rhi = rlo + s;
      float a = ang[p];
      float ca, sa;
      __sincosf(a, &sa, &ca);
      float lo = m[rlo], hi = m[rhi];
      m[rlo] = ca * lo + sa * hi;
      m[rhi] = -sa * lo + ca * hi;
    }
  }

  float* out = mats + ((size_t)b * 2 + t) * 256;
#pragma unroll
  for (int r = 0; r < 16; ++r) out[r * 16 + j] = m[r];
}

// ---------------------------------------------------------------------------
// Kernel 2: streaming WMMA kernel.
// 256 threads = 8 waves. Each wave processes TWO 16-col tiles per iteration
// (32 columns, b64 per lane: even cols -> tile A, odd cols -> tile B), so
// every half-wave request is a full 128B line.
// gridDim.y = block-row b; gridDim.x = 1024-column spans (4 iters of 256).
// ---------------------------------------------------------------------------
#define WAVES_PER_WG 8
#define TILE_ITERS   4
#define ITER_COLS    (WAVES_PER_WG * 32)                // 256
#define SPAN_COLS    (ITER_COLS * TILE_ITERS)           // 1024

__global__ void __launch_bounds__(256)
butterfly_wmma_main(const float* __restrict__ data,
                    const float* __restrict__ biases,
                    const int*   __restrict__ idx_in,
                    const int*   __restrict__ idx_out,
                    const float* __restrict__ mats,
                    float*       __restrict__ out,
                    int B) {
  __shared__ v2f smem[WAVES_PER_WG][16][17];   // per-wave C->B relayout (f32x2)

  const int b    = blockIdx.y;
  const int lane = threadIdx.x & 31;
  const int wave = threadIdx.x >> 5;
  const int hb   = lane >> 4;        // lane half (0 or 1)
  const int r16  = lane & 15;

  // ---- A fragments: M_in / M_out, 4 K-chunks of 16x4 each.
  // A layout (32-bit, 16x4): lane m holds A[m][q*4 + 2*half + {0,1}].
  const float* Min  = mats + (size_t)b * 512;
  const float* Mout = Min + 256;
  v2f aIn[4], aOut[4];
#pragma unroll
  for (int q = 0; q < 4; ++q) {
    const int k = q * 4 + 2 * hb;
    aIn[q]  = *(const v2f*)(Min  + r16 * 16 + k);
    aOut[q] = *(const v2f*)(Mout + r16 * 16 + k);
  }

  // ---- gathered row base pointers (precomputed once; inner loop uses
  // compile-time offsets that fold into the load/store immediate).
  // B-fragment rows k = q*4 + v + 2*half; C-fragment rows m = v + 8*half.
  const int col0 = blockIdx.x * SPAN_COLS + wave * 32 + 2 * r16;

  const float* pin[4][2];
#pragma unroll
  for (int q = 0; q < 4; ++q)
#pragma unroll
    for (int v = 0; v < 2; ++v) {
      const unsigned row = (unsigned)idx_in[b * 16 + q * 4 + v + 2 * hb];
      pin[q][v] = data + (size_t)row * (unsigned)B + col0;
    }

  float* pout[8];
#pragma unroll
  for (int v = 0; v < 8; ++v) {
    const unsigned row = (unsigned)idx_out[b * 16 + v + 8 * hb];
    pout[v] = out + (size_t)row * (unsigned)B + col0;
  }

  // ---- biases for rows 0..7 of this block (wave-uniform -> scalar loads)
  float bia[8];
#pragma unroll
  for (int v = 0; v < 8; ++v) bia[v] = biases[b * NUM_ACT + v];

#pragma unroll
  for (int it = 0; it < TILE_ITERS; ++it) {
    const int colbase = blockIdx.x * SPAN_COLS + it * ITER_COLS + wave * 32;
    if (colbase >= B) continue;          // wave-uniform: EXEC stays all-ones
    const int off = it * ITER_COLS;      // compile-time after unroll

    // ---- load x as b64 per lane: .x -> tile A (even cols), .y -> tile B
    v2f xfA[4], xfB[4];
#pragma unroll
    for (int q = 0; q < 4; ++q) {
      v2f d0 = *(const v2f*)(pin[q][0] + off);
      v2f d1 = *(const v2f*)(pin[q][1] + off);
      xfA[q] = (v2f){d0.x, d1.x};
      xfB[q] = (v2f){d0.y, d1.y};
    }

    // ---- stage 1: y = M_in * x   (4x WMMA per tile)
    v8f accA = {0.f, 0.f, 0.f, 0.f, 0.f, 0.f, 0.f, 0.f};
    v8f accB = {0.f, 0.f, 0.f, 0.f, 0.f, 0.f, 0.f, 0.f};
#pragma unroll
    for (int q = 0; q < 4; ++q) {
      accA = __builtin_amdgcn_wmma_f32_16x16x4_f32(
          false, aIn[q], false, xfA[q], (short)0, accA, false, false);
      accB = __builtin_amdgcn_wmma_f32_16x16x4_f32(
          false, aIn[q], false, xfB[q], (short)0, accB, false, false);
    }

    // ---- fused bias + smooth ReLU on rows 0..7 (== lanes 0..15).
    // Raw v_sqrt_f32 (approx) instead of the expanded IEEE sequence.
#pragma unroll
    for (int v = 0; v < 8; ++v) {
      float yA = accA[v], yB = accB[v];
      float xaA = yA + bia[v];
      float xaB = yB + bia[v];
      float sqA = __builtin_amdgcn_sqrtf(xaA * xaA + CURV2);
      float sqB = __builtin_amdgcn_sqrtf(xaB * xaB + CURV2);
      float acA = 0.5f * xaA + 0.5f * sqA;
      float acB = 0.5f * xaB + 0.5f * sqB;
      accA[v] = (hb == 0) ? acA : yA;    // v_cndmask, no EXEC divergence
      accB[v] = (hb == 0) ? acB : yB;
    }

    // ---- relayout C-fragment -> B-fragment via per-wave LDS (b64 ops,
    // intra-wave DS is in-order so no barrier needed)
#pragma unroll
    for (int v = 0; v < 8; ++v)
      smem[wave][v + 8 * hb][r16] = (v2f){accA[v], accB[v]};

    v2f yfA[4], yfB[4];
#pragma unroll
    for (int q = 0; q < 4; ++q) {
      v2f t0 = smem[wave][q * 4 + 0 + 2 * hb][r16];
      v2f t1 = smem[wave][q * 4 + 1 + 2 * hb][r16];
      yfA[q] = (v2f){t0.x, t1.x};
      yfB[q] = (v2f){t0.y, t1.y};
    }

    // ---- stage 2: z = M_out * y
    v8f zacA = {0.f, 0.f, 0.f, 0.f, 0.f, 0.f, 0.f, 0.f};
    v8f zacB = {0.f, 0.f, 0.f, 0.f, 0.f, 0.f, 0.f, 0.f};
#pragma unroll
    for (int q = 0; q < 4; ++q) {
      zacA = __builtin_amdgcn_wmma_f32_16x16x4_f32(
          false, aOut[q], false, yfA[q], (short)0, zacA, false, false);
      zacB = __builtin_amdgcn_wmma_f32_16x16x4_f32(
          false, aOut[q], false, yfB[q], (short)0, zacB, false, false);
    }

    // ---- scatter back as b64 (idx_out covers all rows in this workload)
#pragma unroll
    for (int v = 0; v < 8; ++v)
      *(v2f*)(pout[v] + off) = (v2f){zacA[v], zacB[v]};
  }
}

// ---------------------------------------------------------------------------
// Host side
// ---------------------------------------------------------------------------
extern "C" void kernel_launch(void* const* d_in, const int* in_sizes, int n_in,
                              void* d_out, int out_size, void* d_ws, size_t ws_size,
                              hipStream_t stream) {
  const float* data   = (const float*)d_in[0];
  const float* angles = (const float*)d_in[1];
  const float* biases = (const float*)d_in[2];
  const int*   idxin  = (const int*)d_in[3];
  const int*   idxout = (const int*)d_in[4];
  float* out = (float*)d_out;

  const int N  = in_sizes[3];                  // 4096
  const int B  = (int)((long)in_sizes[0] / N); // 8192
  const int nb = N / COL_BLOCK;                // 256

  float* mats = (float*)d_ws;                  // nb * 2 * 256 floats (512 KB)

  butterfly_build_mats<<<nb, 32, 0, stream>>>(angles, mats, nb);

  dim3 grid((B + SPAN_COLS - 1) / SPAN_COLS, nb);
  butterfly_wmma_main<<<grid, WAVES_PER_WG * 32, 0, stream>>>(
      data, biases, idxin, idxout, mats, out, B);
}